// BEV2Cyl_36567351558596
// MI455X (gfx1250) — compile-verified
//
#include <hip/hip_runtime.h>
#include <math.h>
#include <stdint.h>

// BEV (4,128,512,512) -> cylindrical (4,128,64,2048): bilinear sample on a
// fixed circle (coords depend only on w), broadcast across H_C=64 rows.
// Store-bound: 256MB output => ~11us floor at 23.3 TB/s.
//
// CDNA5 plan: one block per (b,c). 8 waves compute the 2048-float column into
// LDS (hardware v_sin/v_cos trig, 4-corner gather), then ONE Tensor Data Mover
// op (tensor_store_from_lds, iterate mode) streams the row to global memory
// 64 times: lds_addr_increment=0, global_addr_increment=W_C, iterate_count=63.
// The H-broadcast is done by the DMA engine, not by per-wave store issue.

typedef float        v4f   __attribute__((ext_vector_type(4)));
typedef unsigned int u32x4 __attribute__((ext_vector_type(4)));
typedef int          i32x4 __attribute__((ext_vector_type(4)));
typedef int          i32x8 __attribute__((ext_vector_type(8)));

namespace {
constexpr int B   = 4;
constexpr int C   = 128;
constexpr int H_B = 512;
constexpr int W_B = 512;
constexpr int H_C = 64;
constexpr int W_C = 2048;
constexpr int TPB = 256;                 // 8 wave32
constexpr int WPT = W_C / TPB;           // 8 column values per thread
constexpr float PI_F = 3.14159265358979323846f;
}

__global__ __launch_bounds__(TPB)
void bev2cyl_tdm_kernel(const float* __restrict__ bev, float* __restrict__ out) {
  __shared__ float srow[W_C];            // 8KB: one (b,c) column

  const int bc = blockIdx.x;             // bc = b*C + c
  const float* __restrict__ plane = bev + (size_t)bc * (size_t)(H_B * W_B);

  // ---- compute the 2048-value column cooperatively into LDS ----
  #pragma unroll
  for (int k = 0; k < WPT; ++k) {
    const int   w   = threadIdx.x + k * TPB;      // coalesced, conflict-free
    const float phi = -PI_F + (2.0f * PI_F / (float)(W_C - 1)) * (float)w;
    // |phi| <= pi: hardware v_cos/v_sin domain, no libm range reduction
    const float xg  = 50.0f * __cosf(phi);
    const float yg  = 50.0f * __sinf(phi);
    const float x   = (xg + 50.0f) * ((float)(W_B - 1) / 100.0f);
    const float y   = (50.0f - yg) * ((float)(H_B - 1) / 100.0f);
    const float xf  = floorf(x);
    const float yf  = floorf(y);
    const int   x0  = (int)xf, x1 = x0 + 1;
    const int   y0  = (int)yf, y1 = y0 + 1;
    const float wx1 = x - xf, wx0 = 1.0f - wx1;
    const float wy1 = y - yf, wy0 = 1.0f - wy1;

    // gfx1250 global_prefetch: pre-touch the two corner rows
    {
      const int xc = min(max(x0, 0), W_B - 1);
      const int yc = min(max(y0, 0), H_B - 1);
      __builtin_prefetch(plane + yc * W_B + xc, 0, 3);
      __builtin_prefetch(plane + min(yc + 1, H_B - 1) * W_B + xc, 0, 3);
    }

    auto corner = [&](int xi, int yi, float wgt) -> float {
      const bool valid = (xi >= 0) & (xi < W_B) & (yi >= 0) & (yi < H_B);
      const int  xc = min(max(xi, 0), W_B - 1);
      const int  yc = min(max(yi, 0), H_B - 1);
      const float v = plane[yc * W_B + xc];
      return valid ? v * wgt : 0.0f;
    };

    srow[w] = corner(x0, y0, wx0 * wy0) + corner(x1, y0, wx1 * wy0) +
              corner(x0, y1, wx0 * wy1) + corner(x1, y1, wx1 * wy1);
  }

  __syncthreads();                        // column visible in LDS

#if __has_builtin(__builtin_amdgcn_tensor_store_from_lds)
  // ---- one TDM store per (b,c): iterate the row 64 times into global ----
  if (threadIdx.x == 0) {
    const uint64_t ga = (uint64_t)(uintptr_t)(out + (size_t)bc * (size_t)(H_C * W_C));
    const uint32_t lds_off = (uint32_t)(uintptr_t)(void*)&srow[0];

    // D# group 0: count=1, lds_addr, global_addr[56:0], type=2
    u32x4 g0;
    g0[0] = 1u;
    g0[1] = lds_off;
    g0[2] = (uint32_t)ga;
    g0[3] = (uint32_t)((ga >> 32) & 0x01FFFFFFull) | (2u << 30);

    // D# group 1: data_size=4B(code 2), iterate_enable=1,
    // tensor_dim0=W_C, tensor_dim1=1, tile_dim0=W_C, tensor_dim0_stride=W_C
    i32x8 g1;
    g1[0] = (int)((2u << 16) | (1u << 19));      // data_size | iterate_enable
    g1[1] = (int)((uint32_t)W_C << 16);          // tensor_dim0[15:0] @ bits 63:48
    g1[2] = (int)(1u << 16);                     // dim0 hi=0; tensor_dim1[15:0]=1
    g1[3] = (int)((uint32_t)W_C << 16);          // dim1 hi=0; tile_dim0=W_C
    g1[4] = 0;                                   // tile_dim1=0, tile_dim2=0
    g1[5] = (int)W_C;                            // tensor_dim0_stride lo32
    g1[6] = 0;                                   // stride0 hi; stride1 lo
    g1[7] = 0;

    // D# group 2 (iterate mode): tensor_dim2=1, lds_addr_increment=0,
    // global_addr_increment=W_C elements (one output row), iterate_count=63
    i32x4 g2;
    g2[0] = 1;
    g2[1] = 0;
    g2[2] = (int)W_C;
    g2[3] = (int)(63u << 16);

    i32x4 g3 = {0, 0, 0, 0};

#if defined(__clang_major__) && (__clang_major__ >= 23)
    i32x8 g4 = {0, 0, 0, 0, 0, 0, 0, 0};
    __builtin_amdgcn_tensor_store_from_lds(g0, g1, g2, g3, g4, 0);
#else
    __builtin_amdgcn_tensor_store_from_lds(g0, g1, g2, g3, 0);
#endif
    __builtin_amdgcn_s_wait_tensorcnt(0);
  }
#else
  // ---- fallback: per-wave non-temporal stores of the broadcast rows ----
  float vals[WPT];
  #pragma unroll
  for (int k = 0; k < WPT; ++k) vals[k] = srow[threadIdx.x + k * TPB];

  float* __restrict__ obase = out + (size_t)bc * (size_t)(H_C * W_C);
  for (int h = 0; h < H_C; ++h) {
    float* __restrict__ orow = obase + (size_t)h * W_C;
    #pragma unroll
    for (int k = 0; k < WPT; ++k)
      __builtin_nontemporal_store(vals[k], orow + threadIdx.x + k * TPB);
  }
#endif
}

extern "C" void kernel_launch(void* const* d_in, const int* /*in_sizes*/,
                              int /*n_in*/, void* d_out, int /*out_size*/,
                              void* /*d_ws*/, size_t /*ws_size*/,
                              hipStream_t stream) {
  const float* bev = (const float*)d_in[0];
  float*       out = (float*)d_out;

  dim3 grid(B * C);   // 512 blocks: one (b,c) tile per block
  dim3 block(TPB);
  hipLaunchKernelGGL(bev2cyl_tdm_kernel, grid, block, 0, stream, bev, out);
}